// RGCNLayer_558345748775
// MI455X (gfx1250) — compile-verified
//
#include <hip/hip_runtime.h>

typedef float v2f __attribute__((ext_vector_type(2)));
typedef float v8f __attribute__((ext_vector_type(8)));

#define DDIM 128        // D_IN == D_OUT == 128
#define PSTR 160        // LDS row stride in floats for one K-pair row (64 cols * 2 + 32 pad)
                        // 160 % 64 == 32 -> half-wave kp+1 readers land on banks 32..63

// ---------------------------------------------------------------------------
// Out[z] = X @ W[z] (+ bias), X:[nRows,128] row-major, W:[128,128] row-major.
// Block = 256 threads = 8 waves. Each wave computes a 16(M) x 64(N) tile via
// V_WMMA_F32_16X16X4_F32, so a block covers 128M x 64N.
// The W tile is staged in LDS K-pair-interleaved: Wt[kp*PSTR + 2n + j] =
// W[2kp + j][nb + n], so each B operand (K,K+1 for one column) is a single
// aligned ds_load_b64 straight into an even VGPR pair (no repacking movs).
// ---------------------------------------------------------------------------
__global__ void __launch_bounds__(256)
gemm16x64(const float* __restrict__ X,
          const float* __restrict__ Wbase, size_t wStride,
          const float* __restrict__ bias,
          float* __restrict__ OutBase, size_t outStride,
          int nRows)
{
    __shared__ float Wt[64 * PSTR];     // 40 KB

    const float* W   = Wbase  + (size_t)blockIdx.z * wStride;
    float*       Out = OutBase + (size_t)blockIdx.z * outStride;
    const int    nb  = blockIdx.y * 64;

    // Cooperative load: 64 K-pairs x 64 cols.
    for (int i = threadIdx.x; i < 64 * 64; i += 256) {
        int kp = i >> 6;      // 0..63
        int nc = i & 63;      // 0..63
        Wt[kp * PSTR + 2 * nc + 0] = W[(2 * kp + 0) * DDIM + nb + nc];
        Wt[kp * PSTR + 2 * nc + 1] = W[(2 * kp + 1) * DDIM + nb + nc];
    }
    __syncthreads();

    const int wave   = threadIdx.x >> 5;
    const int lane   = threadIdx.x & 31;
    const int lane16 = lane & 15;
    const int koff   = (lane < 16) ? 0 : 2;   // hi half-wave holds K+2,K+3
    const int kpadd  = (lane < 16) ? 0 : 1;   // same, in K-pair units
    const int mTile  = blockIdx.x * 8 + wave;
    const int mBase  = mTile * 16;
    if (mBase >= nRows) return;               // wave-uniform exit (EXEC stays all-ones)

    v8f c0 = {}, c1 = {}, c2 = {}, c3 = {};
    const float* xrow = X + (size_t)(mBase + lane16) * DDIM + koff;
    const float* wl   = Wt + (size_t)kpadd * PSTR + 2 * lane16;

    for (int k = 0; k < DDIM; k += 4) {
        // A 16x4 fp32 tile: lane<16 -> (K,K+1), lane>=16 -> (K+2,K+3), row M=lane%16
        v2f a = *(const v2f*)(xrow + k);

        const float* wk = wl + (k >> 1) * PSTR;
        v2f b0 = *(const v2f*)(wk +  0);
        v2f b1 = *(const v2f*)(wk + 32);
        v2f b2 = *(const v2f*)(wk + 64);
        v2f b3 = *(const v2f*)(wk + 96);

        c0 = __builtin_amdgcn_wmma_f32_16x16x4_f32(false, a, false, b0, (short)0, c0, false, false);
        c1 = __builtin_amdgcn_wmma_f32_16x16x4_f32(false, a, false, b1, (short)0, c1, false, false);
        c2 = __builtin_amdgcn_wmma_f32_16x16x4_f32(false, a, false, b2, (short)0, c2, false, false);
        c3 = __builtin_amdgcn_wmma_f32_16x16x4_f32(false, a, false, b3, (short)0, c3, false, false);
    }

    // C/D layout: VGPR v holds M = v + (lane>=16 ? 8 : 0), N = lane%16
    const int rowOff = (lane < 16) ? 0 : 8;
    const float bv0 = bias ? bias[nb + lane16 +  0] : 0.0f;
    const float bv1 = bias ? bias[nb + lane16 + 16] : 0.0f;
    const float bv2 = bias ? bias[nb + lane16 + 32] : 0.0f;
    const float bv3 = bias ? bias[nb + lane16 + 48] : 0.0f;
    for (int v = 0; v < 8; ++v) {
        int row = mBase + rowOff + v;
        float* o = Out + (size_t)row * DDIM + nb + lane16;
        o[ 0] = c0[v] + bv0;
        o[16] = c1[v] + bv1;
        o[32] = c2[v] + bv2;
        o[48] = c3[v] + bv3;
    }
}

// ---------------------------------------------------------------------------
__global__ void __launch_bounds__(256)
zero_f32(float* __restrict__ p, size_t n)
{
    size_t i = (size_t)blockIdx.x * 256 + threadIdx.x;
    if (i < n) p[i] = 0.0f;
}

// deg[r*N + dst] += 1 for every edge
__global__ void __launch_bounds__(256)
count_deg(const int* __restrict__ dstIdx, const int* __restrict__ etype,
          float* __restrict__ deg, int nEdges, int nNodes)
{
    int e = blockIdx.x * 256 + threadIdx.x;
    if (e < nEdges)
        atomicAdd(&deg[(size_t)etype[e] * nNodes + dstIdx[e]], 1.0f);
}

// One wave per edge: out[dst] += xw[rel, src] / max(deg[rel,dst],1)
// Each lane handles 4 consecutive output features (float4 gather + 4 atomics).
__global__ void __launch_bounds__(256)
scatter_msgs(const int* __restrict__ srcIdx, const int* __restrict__ dstIdx,
             const int* __restrict__ etype,
             const float* __restrict__ xw, size_t xwStride,
             const float* __restrict__ deg,
             float* __restrict__ out, int nEdges, int nNodes, int relFilter)
{
    int e = blockIdx.x * 8 + (threadIdx.x >> 5);
    if (e >= nEdges) return;
    int r = etype[e];
    if (relFilter >= 0 && r != relFilter) return;
    int s = srcIdx[e];
    int d = dstIdx[e];
    float inv = 1.0f / fmaxf(deg[(size_t)r * nNodes + d], 1.0f);
    int lane = threadIdx.x & 31;

    const float4 m = *((const float4*)(xw + (size_t)r * xwStride + (size_t)s * DDIM) + lane);
    float* o = out + (size_t)d * DDIM + lane * 4;
    atomicAdd(o + 0, m.x * inv);
    atomicAdd(o + 1, m.y * inv);
    atomicAdd(o + 2, m.z * inv);
    atomicAdd(o + 3, m.w * inv);
}

// ---------------------------------------------------------------------------
extern "C" void kernel_launch(void* const* d_in, const int* in_sizes, int n_in,
                              void* d_out, int out_size, void* d_ws, size_t ws_size,
                              hipStream_t stream)
{
    const float* x     = (const float*)d_in[0];
    const int*   eidx  = (const int*)  d_in[1];   // [2, E]: src row then dst row
    const int*   etype = (const int*)  d_in[2];   // [E]
    const float* relW  = (const float*)d_in[3];   // [R,128,128]
    const float* selfW = (const float*)d_in[4];   // [128,128]
    const float* bias  = (const float*)d_in[5];   // [128]
    float*       out   = (float*)d_out;

    const int nNodes = in_sizes[0] / DDIM;
    const int nEdges = in_sizes[2];
    const int nRel   = in_sizes[3] / (DDIM * DDIM);

    const int* srcIdx = eidx;
    const int* dstIdx = eidx + nEdges;

    // workspace layout: [deg: R*N floats][xw buffer]
    float* deg = (float*)d_ws;
    const size_t degElems   = (size_t)nRel * nNodes;
    const size_t degAligned = (degElems * sizeof(float) + 255) & ~(size_t)255;
    float* xw = (float*)((char*)d_ws + degAligned);
    const size_t avail = (ws_size > degAligned) ? ws_size - degAligned : 0;

    const size_t xwRelStride = (size_t)nNodes * DDIM;                  // elems per relation
    const bool   full = avail >= (size_t)nRel * xwRelStride * sizeof(float);

    // 1) degree table
    zero_f32<<<dim3((unsigned)((degElems + 255) / 256)), 256, 0, stream>>>(deg, degElems);
    count_deg<<<dim3((unsigned)((nEdges + 255) / 256)), 256, 0, stream>>>(dstIdx, etype, deg,
                                                                          nEdges, nNodes);

    // 2) self-loop: out = x @ W_self + bias (writes every output element)
    dim3 gg((unsigned)((nNodes + 127) / 128), DDIM / 64, 1);
    gemm16x64<<<gg, 256, 0, stream>>>(x, selfW, 0, bias, out, 0, nNodes);

    // 3) relation GEMMs + edge scatter
    if (full) {
        dim3 gr(gg.x, gg.y, (unsigned)nRel);
        gemm16x64<<<gr, 256, 0, stream>>>(x, relW, (size_t)DDIM * DDIM, nullptr,
                                          xw, xwRelStride, nNodes);
        scatter_msgs<<<dim3((unsigned)((nEdges + 7) / 8)), 256, 0, stream>>>(
            srcIdx, dstIdx, etype, xw, xwRelStride, deg, out, nEdges, nNodes, -1);
    } else {
        for (int r = 0; r < nRel; ++r) {
            gemm16x64<<<gg, 256, 0, stream>>>(x, relW + (size_t)r * DDIM * DDIM, 0, nullptr,
                                              xw, 0, nNodes);
            scatter_msgs<<<dim3((unsigned)((nEdges + 7) / 8)), 256, 0, stream>>>(
                srcIdx, dstIdx, etype, xw, 0, deg, out, nEdges, nNodes, r);
        }
    }
}